// TemporalTracker_60705067761963
// MI455X (gfx1250) — compile-verified
//
#include <hip/hip_runtime.h>
#include <hip/hip_bf16.h>

typedef __attribute__((ext_vector_type(2))) float v2f;
typedef __attribute__((ext_vector_type(8))) float v8f;

#define NB    4
#define NT    8192
#define CHUNK 2048          // xt1 points staged per LDS pass (32 KB as float4)

// One 16x16 distance tile: D[M][N] = -2*<x_M,y_N> + |y_N|^2  (same argmin as
// true squared distance; |x_M|^2 is constant along the argmin axis).
__device__ __forceinline__ void tile_update(float4 q, bool hi, v2f A, int n,
                                            float bestv[8], int besti[8])
{
    v2f Bv;
    Bv.x = hi ? q.z : q.x;           // lanes 0-15: K=0 row (x), lanes 16-31: K=2 row (z)
    Bv.y = hi ? q.w : q.y;           // lanes 0-15: K=1 row (y), lanes 16-31: K=3 row (|p|^2)
    const v8f Cz = {};
    v8f d = __builtin_amdgcn_wmma_f32_16x16x4_f32(
        false, A, false, Bv, (short)0, Cz, false, false);
#pragma unroll
    for (int r = 0; r < 8; ++r) {
        if (d[r] < bestv[r]) { bestv[r] = d[r]; besti[r] = n; }
    }
}

// Each wave handles one 16-row tile of xt; 8 waves per block; 512 tiles/batch.
// grid = NB*512/8 = 256 blocks of 256 threads.
__global__ __launch_bounds__(256)
void TemporalTracker_nn_kernel(const float* __restrict__ xt,
                               const float* __restrict__ xt1,
                               float* __restrict__ out)
{
    __shared__ float4 sm[CHUNK];

    const int tid  = threadIdx.x;
    const int lane = tid & 31;
    const int sub  = lane & 15;
    const bool hi  = lane >= 16;
    const int wave = tid >> 5;

    const int mtile = blockIdx.x * 8 + wave;   // 0..2047
    const int b     = mtile >> 9;              // 512 M-tiles per batch
    const int m0    = (mtile & 511) << 4;

    // A operand: 16x4 f32 tile of xt rows m0..m0+15, scaled by -2, with the
    // K=3 slot = 1.0 so B's K=3 row injects |y|^2.
    // Layout: lanes 0-15 hold K=0,1 ; lanes 16-31 hold K=2,3 (M = lane&15).
    const float* pa = xt + ((size_t)b * NT + (size_t)(m0 + sub)) * 3;
    const float ax = pa[0], ay = pa[1], az = pa[2];
    v2f A;
    A.x = -2.0f * (hi ? az : ax);
    A.y = hi ? 1.0f : (-2.0f * ay);

    float bestv[8];
    int   besti[8];
#pragma unroll
    for (int r = 0; r < 8; ++r) { bestv[r] = 3.4e38f; besti[r] = 0; }

    for (int c0 = 0; c0 < NT; c0 += CHUNK) {
        __syncthreads();
        // Cooperative stage: float4 (x, y, z, |p|^2) per xt1 point.
        for (int i = tid; i < CHUNK; i += 256) {
            const float* p = xt1 + ((size_t)b * NT + (size_t)(c0 + i)) * 3;
            const float x = p[0], y = p[1], z = p[2];
            sm[i] = make_float4(x, y, z, x * x + y * y + z * z);
        }
        __syncthreads();

        // Software-pipelined B-tile reads: prefetch tile t+1, compute tile t.
        float4 q = sm[sub];
#pragma unroll 2
        for (int t = 0; t < CHUNK - 16; t += 16) {
            const float4 qn = sm[t + 16 + sub];
            tile_update(q, hi, A, c0 + t + sub, bestv, besti);
            q = qn;
        }
        tile_update(q, hi, A, c0 + (CHUNK - 16) + sub, bestv, besti);
    }

    // argmin reduction across the 16 lanes of each half (ties -> lower index),
    // branch-free (no short-circuit) so it lowers to cmp/cndmask.
#pragma unroll
    for (int s = 1; s <= 8; s <<= 1) {
#pragma unroll
        for (int r = 0; r < 8; ++r) {
            const float ov = __shfl_xor(bestv[r], s, 32);
            const int   oi = __shfl_xor(besti[r], s, 32);
            const bool take = (ov < bestv[r]) |
                              ((ov == bestv[r]) & (oi < besti[r]));
            bestv[r] = take ? ov : bestv[r];
            besti[r] = take ? oi : besti[r];
        }
    }

    // Outputs (flat concat of the 5-tuple, all as float).
    if (sub == 0) {
        float* out0 = out;                 // matched_indices_t   [B,NT]
        float* out1 = out + 32768;         // matched_indices_t1  [B,NT]
        float* out2 = out + 65536;         // match_confidences   [B,NT]
        float* out3 = out + 98304;         // trajectories        [B,NT,2,3]
        float* out4 = out + 294912;        // velocities          [B,NT,3]
#pragma unroll
        for (int r = 0; r < 8; ++r) {
            const int m    = m0 + r + (hi ? 8 : 0);
            const int best = besti[r];
            const size_t row = (size_t)b * NT + (size_t)m;

            out0[row] = (float)m;
            out1[row] = (float)best;
            out2[row] = 1.0f;

            const float* pt = xt  + ((size_t)b * NT + (size_t)m) * 3;
            const float* p1 = xt1 + ((size_t)b * NT + (size_t)best) * 3;
            const float x0 = pt[0], y0 = pt[1], z0 = pt[2];
            const float x1 = p1[0], y1 = p1[1], z1 = p1[2];

            float* tr = out3 + row * 6;
            tr[0] = x0; tr[1] = y0; tr[2] = z0;
            tr[3] = x1; tr[4] = y1; tr[5] = z1;

            float* ve = out4 + row * 3;
            ve[0] = x1 - x0; ve[1] = y1 - y0; ve[2] = z1 - z0;
        }
    }
}

extern "C" void kernel_launch(void* const* d_in, const int* in_sizes, int n_in,
                              void* d_out, int out_size, void* d_ws, size_t ws_size,
                              hipStream_t stream) {
    (void)in_sizes; (void)n_in; (void)out_size; (void)d_ws; (void)ws_size;
    const float* xt  = (const float*)d_in[0];
    const float* xt1 = (const float*)d_in[1];
    float* out = (float*)d_out;
    TemporalTracker_nn_kernel<<<dim3(NB * (NT / 16) / 8), dim3(256), 0, stream>>>(xt, xt1, out);
}